// DSNTDistanceDoubleLoss_55035710931290
// MI455X (gfx1250) — compile-verified
//
#include <hip/hip_runtime.h>
#include <math.h>

// Problem geometry (fixed by the reference): B=32, C=2, H=W=512.
#define THREADS   256
#define HDIM      512
#define WDIM      512
#define NSLAB     (HDIM * WDIM)        // 262144 elements per (b,c) slab
#define SPLIT     16                   // blocks per slab
#define CHUNK     (NSLAB / SPLIT)      // 16384 elements per block
#define VEC       4                    // float4 per thread per tile
#define TILE      (THREADS * VEC)      // 1024 elements (4 KB) per tile
#define NTILES    (CHUNK / TILE)       // 16 tiles per block
#define NSLABS    64                   // B*C
#define NPART     (NSLABS * SPLIT)     // 1024 partial records

// ---- CDNA5 async global->LDS copy (ASYNCcnt-tracked), 16B per lane ----
__device__ __forceinline__ void async_load_f4(const float* gsrc, const float* ldst) {
  unsigned loff = (unsigned)(unsigned long long)ldst;   // LDS byte address (low 32 bits)
  asm volatile("global_load_async_to_lds_b128 %0, %1, off"
               :: "v"(loff), "v"(gsrc) : "memory");
}
__device__ __forceinline__ void wait_async_le1() {
  asm volatile("s_wait_asynccnt 0x1" ::: "memory");
}
__device__ __forceinline__ void wait_async_le0() {
  asm volatile("s_wait_asynccnt 0x0" ::: "memory");
}

// Kernel 1: per-block partial reductions over one CHUNK of one (b,c) slab.
// Produces: sums[p] = {sum(e), sum(e*x), sum(e*y), unused}, maxv[p], maxi[p].
__global__ void dsnt_partial_kernel(const float* __restrict__ in,
                                    const float* __restrict__ tgt,
                                    float4* __restrict__ sums,
                                    float*  __restrict__ maxv,
                                    int*    __restrict__ maxi) {
  __shared__ __align__(16) float tile[2][TILE];
  __shared__ float rs[THREADS], rsx[THREADS], rsy[THREADS], rmv[THREADS];
  __shared__ int   rmi[THREADS];

  const int blk  = blockIdx.x;
  const int sc   = blk / SPLIT;          // slab = b*C + c
  const int part = blk % SPLIT;
  const int t    = threadIdx.x;
  const long base = (long)sc * NSLAB + (long)part * CHUNK;
  const float* gin = in  + base;
  const float* gtg = tgt + base;

  // Prefetch tile 0 into LDS buffer 0 (async, per-wave counter).
  async_load_f4(gin + t * VEC, &tile[0][t * VEC]);

  float s = 0.0f, sx = 0.0f, sy = 0.0f;
  float mval = -INFINITY;
  int   midx = 0;

  const float inv = 1.0f / 512.0f;

  for (int it = 0; it < NTILES; ++it) {
    const int buf = it & 1;
    // Prefetch next tile, then wait for the current one (in-order completion).
    if (it + 1 < NTILES) {
      async_load_f4(gin + (long)(it + 1) * TILE + t * VEC, &tile[buf ^ 1][t * VEC]);
      wait_async_le1();
    } else {
      wait_async_le0();
    }

    // Target: plain coalesced b128 load (argmax path, no staging needed).
    const float4 tv = *(const float4*)(gtg + (long)it * TILE + t * VEC);
    // Input tile from LDS (each thread reads back exactly what it loaded).
    const float4 v  = *(const float4*)&tile[buf][t * VEC];

    const int jbase = part * CHUNK + it * TILE + t * VEC;
    const float vv[4] = {v.x, v.y, v.z, v.w};
    const float tt[4] = {tv.x, tv.y, tv.z, tv.w};
#pragma unroll
    for (int k = 0; k < 4; ++k) {
      const int j  = jbase + k;
      const float e  = __expf(vv[k]);
      const float xw = (float)((j & (WDIM - 1)) + 1) * inv;
      const float yh = (float)((j >> 9) + 1) * inv;
      s += e;
      sx = fmaf(e, xw, sx);
      sy = fmaf(e, yh, sy);
      const float tk = tt[k];
      if (tk > mval || (tk == mval && j < midx)) { mval = tk; midx = j; }
    }
  }

  // Block tree reduction (256 threads).
  rs[t] = s; rsx[t] = sx; rsy[t] = sy; rmv[t] = mval; rmi[t] = midx;
  __syncthreads();
  for (int off = THREADS / 2; off > 0; off >>= 1) {
    if (t < off) {
      rs[t]  += rs[t + off];
      rsx[t] += rsx[t + off];
      rsy[t] += rsy[t + off];
      const float ov = rmv[t + off];
      const int   oi = rmi[t + off];
      if (ov > rmv[t] || (ov == rmv[t] && oi < rmi[t])) { rmv[t] = ov; rmi[t] = oi; }
    }
    __syncthreads();
  }
  if (t == 0) {
    sums[blk] = make_float4(rs[0], rsx[0], rsy[0], 0.0f);
    maxv[blk] = rmv[0];
    maxi[blk] = rmi[0];
  }
}

// Kernel 2: one wave32; lane b handles batch b (both channels), then wave-reduce.
__global__ void dsnt_final_kernel(const float4* __restrict__ sums,
                                  const float*  __restrict__ maxv,
                                  const int*    __restrict__ maxi,
                                  float* __restrict__ out) {
  const int b = threadIdx.x;             // 0..31
  const float inv = 1.0f / 512.0f;
  float px[2], py[2], tx[2], ty[2];
  float total = 0.0f;

#pragma unroll
  for (int c = 0; c < 2; ++c) {
    const int sc = b * 2 + c;
    float s = 0.0f, sx = 0.0f, sy = 0.0f;
    float mv = -INFINITY; int mi = 0;
    for (int p = 0; p < SPLIT; ++p) {
      const int r = sc * SPLIT + p;
      const float4 q = sums[r];
      s += q.x; sx += q.y; sy += q.z;
      const float v = maxv[r]; const int i = maxi[r];
      if (v > mv || (v == mv && i < mi)) { mv = v; mi = i; }
    }
    px[c] = sx / s;
    py[c] = sy / s;
    tx[c] = (float)((mi & (WDIM - 1)) + 1) * inv;
    ty[c] = (float)((mi >> 9) + 1) * inv;
    const float dx = tx[c] - px[c], dy = ty[c] - py[c];
    total += sqrtf(dx * dx + dy * dy);
  }

  const float pdx = px[0] - px[1], pdy = py[0] - py[1];
  const float tdx = tx[0] - tx[1], tdy = ty[0] - ty[1];
  const float pd = sqrtf(pdx * pdx + pdy * pdy);
  const float td = sqrtf(tdx * tdx + tdy * tdy);
  total += fabsf(pd - td);

  // wave32 reduction
#pragma unroll
  for (int off = 16; off > 0; off >>= 1)
    total += __shfl_xor(total, off, 32);

  if (b == 0) out[0] = total * (1.0f / 32.0f);
}

extern "C" void kernel_launch(void* const* d_in, const int* in_sizes, int n_in,
                              void* d_out, int out_size, void* d_ws, size_t ws_size,
                              hipStream_t stream) {
  (void)in_sizes; (void)n_in; (void)out_size; (void)ws_size;
  const float* input  = (const float*)d_in[0];
  const float* target = (const float*)d_in[1];
  float* out = (float*)d_out;

  char* ws = (char*)d_ws;
  float4* sums = (float4*)ws;                                        // NPART * 16 B
  float*  maxv = (float*)(ws + (size_t)NPART * sizeof(float4));      // NPART * 4 B
  int*    maxi = (int*)  (ws + (size_t)NPART * (sizeof(float4) + sizeof(float)));

  dsnt_partial_kernel<<<NPART, THREADS, 0, stream>>>(input, target, sums, maxv, maxi);
  dsnt_final_kernel<<<1, 32, 0, stream>>>(sums, maxv, maxi, out);
}